// ConvNL_46720654245937
// MI455X (gfx1250) — compile-verified
//
#include <hip/hip_runtime.h>
#include <hip/hip_bf16.h>
#include <math.h>

// ---------------- problem constants ----------------
#define BB   16
#define CIN  64
#define COUT 128
#define HH   512
#define WW   64
#define KTOT (CIN*9)          // 576
#define NPIX (BB*HH*WW)       // per-channel element count = 524288
#define EPSV 1e-5f
#define STRIP 32              // conv output width per block (2 WMMA tiles)

typedef float v2f __attribute__((ext_vector_type(2)));
typedef float v8f __attribute__((ext_vector_type(8)));

// ---------------- ws layout (floats) ----------------
#define XM_OFF   0u                 // 16*128*512        = 1048576
#define F_OFF    1048576u           // 16*512*512        = 4194304
#define Y_OFF    5242880u           // 1048576
#define Z_OFF    6291456u           // 1048576
#define YO_OFF   7340032u           // 1048576
#define STAT_OFF 8388608u           // 1024 floats of stats
#define S_CSUM1   0
#define S_CSQ1    128
#define S_SC1     256
#define S_BI1     384
#define S_BSUM    512
#define S_BSQ     528
#define S_MU      544
#define S_RS      560
#define S_CSUM2   576
#define S_CSQ2    704
#define S_SC2     832
#define S_BI2     960
#define STAT_N    1024

// ==================================================================
// K0: zero the stats region (must re-zero every call: atomics accumulate)
// ==================================================================
__global__ void zero_stats(float* stat) {
    for (int i = threadIdx.x; i < STAT_N; i += 256) stat[i] = 0.0f;
}

// ==================================================================
// K1: 3x3 reflect conv as implicit GEMM, V_WMMA_F32_16X16X4_F32.
//     grid = B*H*(W/32) blocks of 256 (8 waves; wave = one Cout slice,
//     TWO 16x16 w-tiles sharing one A fragment -> 2 WMMA per weight load).
//
//     Stage 1: raw reflect patch loaded with GLOBAL_LOAD_ASYNC_TO_LDS_B32
//              (ASYNCcnt path), drained with s_wait_asynccnt.
//     Stage 2: expand to im2col in exact B-fragment order:
//       dword addr = kk*128 + tile*64 + khalf*32 + m*2 + j
//       holds B[k = kk*4 + khalf*2 + j][n = tile*16 + m]
//     => per ds_load_b64, khalf0 lanes hit banks 0..31 and khalf1 lanes
//     hit banks 32..63 (conflict-free); hot loop has constant strides,
//     no divisions.
// ==================================================================
__global__ void __launch_bounds__(256) conv_wmma(const float* __restrict__ x,
                                                 const float* __restrict__ cw,
                                                 float* __restrict__ hraw) {
    __shared__ float raw[CIN * 3 * (STRIP + 2)];   // 26.1 KB [ci][r][col 0..33]
    __shared__ float pe[KTOT * STRIP];             // 73.7 KB B-fragment order
    const int blk = blockIdx.x;
    const int wt  = blk & 1;                 // w strip
    const int bh  = (blk >> 1) & (HH - 1);   // h
    const int bb  = blk >> 10;               // batch
    const int w0  = wt * STRIP;
    const int tid = threadIdx.x;

    // stage 1: async copy raw patch (reflect padded) straight into LDS
    for (int i = tid; i < CIN * 3 * (STRIP + 2); i += 256) {
        int ci  = i / 102;                  // 3*(STRIP+2) = 102
        int rem = i - ci * 102;
        int r   = rem / 34;
        int col = rem - r * 34;
        int hh  = bh + r - 1;
        hh = (hh < 0) ? 1 : (hh > HH - 1 ? HH - 2 : hh);
        int ww = w0 + col - 1;
        ww = (ww < 0) ? 1 : (ww > WW - 1 ? WW - 2 : ww);
        const float* gp = x + (((bb * CIN) + ci) * HH + hh) * WW + ww;
        unsigned lds_off = (unsigned)(size_t)(&raw[i]);   // LDS byte address
        asm volatile("global_load_async_to_lds_b32 %0, %1, off"
                     :: "v"(lds_off), "v"(gp) : "memory");
    }
    asm volatile("s_wait_asynccnt 0x0" ::: "memory");
    __syncthreads();

    // stage 2: expand into WMMA B-fragment order (divisions only here)
    for (int i = tid; i < KTOT * STRIP; i += 256) {
        int kk  = i >> 7;
        int rem = i & 127;
        int t   = (rem >> 6) & 1;
        int kh  = (rem >> 5) & 1;
        int mm  = (rem >> 1) & 15;
        int j   = i & 1;
        int k   = (kk << 2) + (kh << 1) + j;
        int ci  = k / 9;
        int r9  = k - ci * 9;
        int r   = r9 / 3;
        int kw  = r9 - r * 3;
        pe[i] = raw[ci * 102 + r * 34 + kw + (t << 4) + mm];
    }
    __syncthreads();

    const int wave  = tid >> 5;
    const int lane  = tid & 31;
    const int m     = lane & 15;     // A row (Cout) / B col (spatial)
    const int khalf = lane >> 4;     // K pair select
    const int co0   = wave << 4;

    const float* wk    = cw + (co0 + m) * KTOT + (khalf << 1);
    const float* pbase = pe + (khalf << 5) + (m << 1);
    __builtin_prefetch(wk, 0, 0);            // global_prefetch_b8
    __builtin_prefetch(wk + 288, 0, 0);

    v8f acc0 = {}, acc1 = {};
#pragma unroll 2
    for (int kk = 0; kk < KTOT / 4; ++kk) {
        v2f a  = *(const v2f*)(wk + (kk << 2));            // global_load_b64
        v2f b0 = *(const v2f*)(pbase + (kk << 7));         // ds_load_b64
        v2f b1 = *(const v2f*)(pbase + (kk << 7) + 64);    // ds_load_b64
        acc0 = __builtin_amdgcn_wmma_f32_16x16x4_f32(false, a, false, b0,
                                                     (short)0, acc0, false, false);
        acc1 = __builtin_amdgcn_wmma_f32_16x16x4_f32(false, a, false, b1,
                                                     (short)0, acc1, false, false);
    }
    // C/D layout: VGPR v -> M = v + khalf*8, N = m
#pragma unroll
    for (int v = 0; v < 8; ++v) {
        int co = co0 + v + (khalf << 3);
        float* orow = hraw + (((bb * COUT) + co) * HH + bh) * WW + w0 + m;
        orow[0]  = acc0[v];
        orow[16] = acc1[v];
    }
}

// ==================================================================
// K2: per-channel sum / sumsq over hraw (128 ch * 8 chunks blocks)
// ==================================================================
__global__ void __launch_bounds__(256) chan_stats(const float* __restrict__ d,
                                                  float* __restrict__ csum,
                                                  float* __restrict__ csq) {
    __shared__ float red[256];
    const int c     = blockIdx.x >> 3;
    const int chunk = blockIdx.x & 7;
    float s = 0.f, sq = 0.f;
    for (int e = chunk * 65536 + threadIdx.x; e < (chunk + 1) * 65536; e += 256) {
        int bb  = e >> 15;                 // 32768 elems per (b,c) slab
        int off = e & 32767;
        float v = d[((bb * COUT + c) << 15) + off];
        s += v; sq += v * v;
    }
    red[threadIdx.x] = s; __syncthreads();
    for (int st = 128; st > 0; st >>= 1) { if (threadIdx.x < st) red[threadIdx.x] += red[threadIdx.x + st]; __syncthreads(); }
    if (threadIdx.x == 0) atomicAdd(&csum[c], red[0]);
    __syncthreads();
    red[threadIdx.x] = sq; __syncthreads();
    for (int st = 128; st > 0; st >>= 1) { if (threadIdx.x < st) red[threadIdx.x] += red[threadIdx.x + st]; __syncthreads(); }
    if (threadIdx.x == 0) atomicAdd(&csq[c], red[0]);
}

// ==================================================================
// K3: finalize BN params:  scale = g*rsqrt(var+eps), bias = b - mean*scale
// ==================================================================
__global__ void bn_finalize(const float* csum, const float* csq,
                            const float* g, const float* bt,
                            float invN, float* scale, float* bias) {
    int c = threadIdx.x;   // 128 threads
    float mean = csum[c] * invN;
    float var  = csq[c] * invN - mean * mean;
    float sc   = g[c] * rsqrtf(var + EPSV);
    scale[c] = sc;
    bias[c]  = bt[c] - mean * sc;
}

// ==================================================================
// K4: xm[b,c,h] = mean_w relu(bn1(hraw));  fused LN partial sums per batch
// ==================================================================
__global__ void __launch_bounds__(256) xm_kernel(const float* __restrict__ hraw,
                                                 const float* __restrict__ s1,
                                                 const float* __restrict__ b1,
                                                 float* __restrict__ xm,
                                                 float* __restrict__ bsum,
                                                 float* __restrict__ bsq) {
    __shared__ float red[256];
    const int blk = blockIdx.x;
    const int hc  = blk & 1;
    const int c   = (blk >> 1) & (COUT - 1);
    const int b   = blk >> 8;
    const int h   = (hc << 8) + threadIdx.x;
    const float* base = hraw + (((b * COUT) + c) * HH + h) * WW;
    const float sc = s1[c], bi = b1[c];
    float s = 0.f;
#pragma unroll
    for (int w = 0; w < WW; ++w) {
        float v = fmaf(base[w], sc, bi);
        s += fmaxf(v, 0.f);
    }
    const float xv = s * (1.0f / (float)WW);
    xm[((b * COUT) + c) * HH + h] = xv;

    red[threadIdx.x] = xv; __syncthreads();
    for (int st = 128; st > 0; st >>= 1) { if (threadIdx.x < st) red[threadIdx.x] += red[threadIdx.x + st]; __syncthreads(); }
    if (threadIdx.x == 0) atomicAdd(&bsum[b], red[0]);
    __syncthreads();
    red[threadIdx.x] = xv * xv; __syncthreads();
    for (int st = 128; st > 0; st >>= 1) { if (threadIdx.x < st) red[threadIdx.x] += red[threadIdx.x + st]; __syncthreads(); }
    if (threadIdx.x == 0) atomicAdd(&bsq[b], red[0]);
}

// ==================================================================
// K5: layernorm finalize per batch (over C*H = 65536 elems)
// ==================================================================
__global__ void ln_finalize(const float* bsum, const float* bsq, float* mu, float* rs) {
    int b = threadIdx.x;          // 16 threads
    const float invN = 1.0f / (float)(COUT * HH);
    float m = bsum[b] * invN;
    float v = bsq[b] * invN - m * m;
    mu[b] = m;
    rs[b] = rsqrtf(v + EPSV);
}

// ==================================================================
// K6: S[b] = xn^T xn / sqrt(C)   (M=N=512, K=128) — WMMA f32
// ==================================================================
__global__ void __launch_bounds__(256) sgemm_wmma(const float* __restrict__ xm,
                                                  const float* __restrict__ mu,
                                                  const float* __restrict__ rs,
                                                  float* __restrict__ f) {
    const int tid = threadIdx.x, wave = tid >> 5, lane = tid & 31;
    const int b   = blockIdx.x >> 7;
    const int t   = ((blockIdx.x & 127) << 3) + wave;   // 0..1023
    const int i0  = (t >> 5) << 4;
    const int j0  = (t & 31) << 4;
    const int m = lane & 15, khalf = lane >> 4;
    const float mub = mu[b], rsb = rs[b];
    const float* xb = xm + b * COUT * HH;

    v8f acc = {};
#pragma unroll 4
    for (int k = 0; k < COUT; k += 4) {
        const int ka = k + (khalf << 1);
        v2f a, bv;
        a.x  = (xb[ka * HH + i0 + m] - mub) * rsb;
        a.y  = (xb[(ka + 1) * HH + i0 + m] - mub) * rsb;
        bv.x = (xb[ka * HH + j0 + m] - mub) * rsb;
        bv.y = (xb[(ka + 1) * HH + j0 + m] - mub) * rsb;
        acc = __builtin_amdgcn_wmma_f32_16x16x4_f32(false, a, false, bv,
                                                    (short)0, acc, false, false);
    }
    const float scale = 0.08838834764831845f;  // 1/sqrt(128)
    float* fb = f + b * HH * HH;
#pragma unroll
    for (int v = 0; v < 8; ++v) {
        int i = i0 + v + (khalf << 3);
        fb[i * HH + j0 + m] = acc[v] * scale;
    }
}

// ==================================================================
// K7: row softmax over f (8192 rows of 512)
// ==================================================================
__global__ void __launch_bounds__(256) softmax_kernel(float* __restrict__ f) {
    __shared__ float red[256];
    float* row = f + (size_t)blockIdx.x * HH;
    const int tid = threadIdx.x;
    float v0 = row[tid], v1 = row[tid + 256];
    red[tid] = fmaxf(v0, v1); __syncthreads();
    for (int s = 128; s > 0; s >>= 1) { if (tid < s) red[tid] = fmaxf(red[tid], red[tid + s]); __syncthreads(); }
    const float mx = red[0]; __syncthreads();
    float e0 = __expf(v0 - mx), e1 = __expf(v1 - mx);
    red[tid] = e0 + e1; __syncthreads();
    for (int s = 128; s > 0; s >>= 1) { if (tid < s) red[tid] += red[tid + s]; __syncthreads(); }
    const float inv = 1.0f / red[0];
    row[tid] = e0 * inv; row[tid + 256] = e1 * inv;
}

// ==================================================================
// K8: C[b] = A(128x128) @ Bm[b](128x512) + bias   (WMMA f32)
//     normB=1 -> B normalized with (mu,rs) on the fly (B = xn)
// ==================================================================
__global__ void __launch_bounds__(256) gemm128_wmma(const float* __restrict__ A,
                                                    const float* __restrict__ Bm,
                                                    const float* __restrict__ bias,
                                                    const float* __restrict__ mu,
                                                    const float* __restrict__ rs,
                                                    int normB,
                                                    float* __restrict__ C) {
    const int tid = threadIdx.x, wave = tid >> 5, lane = tid & 31;
    const int b  = blockIdx.x >> 5;
    const int t  = ((blockIdx.x & 31) << 3) + wave;     // 0..255
    const int m0 = (t >> 5) << 4;                        // 0..112
    const int n0 = (t & 31) << 4;                        // 0..496
    const int m = lane & 15, khalf = lane >> 4;
    const float mub = normB ? mu[b] : 0.f;
    const float rsb = normB ? rs[b] : 1.f;
    const float* Bb   = Bm + b * COUT * HH;
    const float* Arow = A + (m0 + m) * COUT + (khalf << 1);

    v8f acc = {};
#pragma unroll 4
    for (int k = 0; k < COUT; k += 4) {
        v2f a = *(const v2f*)(Arow + k);
        const int ka = k + (khalf << 1);
        v2f bv;
        bv.x = (Bb[ka * HH + n0 + m] - mub) * rsb;
        bv.y = (Bb[(ka + 1) * HH + n0 + m] - mub) * rsb;
        acc = __builtin_amdgcn_wmma_f32_16x16x4_f32(false, a, false, bv,
                                                    (short)0, acc, false, false);
    }
    float* Cb = C + b * COUT * HH;
#pragma unroll
    for (int v = 0; v < 8; ++v) {
        int mm = m0 + v + (khalf << 3);
        Cb[mm * HH + n0 + m] = acc[v] + bias[mm];
    }
}

// ==================================================================
// K9: z[b][m][h] = sum_k y[b][m][k] * f[b][h][k]  (M=128, N=512, K=512)
// ==================================================================
__global__ void __launch_bounds__(256) zgemm_wmma(const float* __restrict__ y,
                                                  const float* __restrict__ f,
                                                  float* __restrict__ z) {
    const int tid = threadIdx.x, wave = tid >> 5, lane = tid & 31;
    const int b  = blockIdx.x >> 5;
    const int t  = ((blockIdx.x & 31) << 3) + wave;
    const int m0 = (t >> 5) << 4;
    const int n0 = (t & 31) << 4;
    const int m = lane & 15, khalf = lane >> 4;
    const float* Yrow = y + b * COUT * HH + (m0 + m) * HH + (khalf << 1);
    const float* Frow = f + b * HH * HH + (n0 + m) * HH + (khalf << 1);

    v8f acc = {};
#pragma unroll 4
    for (int k = 0; k < HH; k += 4) {
        v2f a  = *(const v2f*)(Yrow + k);
        v2f bv = *(const v2f*)(Frow + k);
        acc = __builtin_amdgcn_wmma_f32_16x16x4_f32(false, a, false, bv,
                                                    (short)0, acc, false, false);
    }
    float* Zb = z + b * COUT * HH;
#pragma unroll
    for (int v = 0; v < 8; ++v) {
        int mm = m0 + v + (khalf << 3);
        Zb[mm * HH + n0 + m] = acc[v];
    }
}

// ==================================================================
// K10: in-place residual  d = relu(bn1(d)) + yout[b,c,h]  + bn2 stats
// ==================================================================
__global__ void __launch_bounds__(256) residual_stats(float* __restrict__ d,
                                                      const float* __restrict__ s1,
                                                      const float* __restrict__ b1,
                                                      const float* __restrict__ yo,
                                                      float* __restrict__ csum,
                                                      float* __restrict__ csq) {
    __shared__ float red[256];
    const int blk   = blockIdx.x;
    const int chunk = blk & 7;
    const int c     = (blk >> 3) & (COUT - 1);
    const int b     = blk >> 10;
    const int slab  = (b * COUT + c) << 15;        // 32768 elems per (b,c)
    const float sc = s1[c], bi = b1[c];
    const float* yrow = yo + (b * COUT + c) * HH;
    float s = 0.f, sq = 0.f;
    for (int e = (chunk << 12) + threadIdx.x; e < ((chunk + 1) << 12); e += 256) {
        float hv = fmaxf(fmaf(d[slab + e], sc, bi), 0.f);
        float v  = hv + yrow[e >> 6];              // broadcast over W
        d[slab + e] = v;
        s += v; sq += v * v;
    }
    red[threadIdx.x] = s; __syncthreads();
    for (int st = 128; st > 0; st >>= 1) { if (threadIdx.x < st) red[threadIdx.x] += red[threadIdx.x + st]; __syncthreads(); }
    if (threadIdx.x == 0) atomicAdd(&csum[c], red[0]);
    __syncthreads();
    red[threadIdx.x] = sq; __syncthreads();
    for (int st = 128; st > 0; st >>= 1) { if (threadIdx.x < st) red[threadIdx.x] += red[threadIdx.x + st]; __syncthreads(); }
    if (threadIdx.x == 0) atomicAdd(&csq[c], red[0]);
}

// ==================================================================
// K11: in-place bn2 + SiLU
// ==================================================================
__global__ void __launch_bounds__(256) silu_apply(float* __restrict__ d,
                                                  const float* __restrict__ s2,
                                                  const float* __restrict__ b2) {
    const int i = blockIdx.x * 256 + threadIdx.x;
    const int c = (i >> 15) & (COUT - 1);
    float v = fmaf(d[i], s2[c], b2[c]);
    d[i] = v / (1.0f + __expf(-v));
}

// ==================================================================
extern "C" void kernel_launch(void* const* d_in, const int* in_sizes, int n_in,
                              void* d_out, int out_size, void* d_ws, size_t ws_size,
                              hipStream_t stream) {
    const float* x      = (const float*)d_in[0];
    const float* conv_w = (const float*)d_in[1];
    const float* bn1_g  = (const float*)d_in[2];
    const float* bn1_b  = (const float*)d_in[3];
    const float* g_w    = (const float*)d_in[4];
    const float* g_b    = (const float*)d_in[5];
    const float* out_w  = (const float*)d_in[6];
    const float* out_b  = (const float*)d_in[7];
    const float* bn2_g  = (const float*)d_in[8];
    const float* bn2_b  = (const float*)d_in[9];

    float* out  = (float*)d_out;          // doubles as h_raw staging buffer
    float* ws   = (float*)d_ws;
    float* xm   = ws + XM_OFF;
    float* f    = ws + F_OFF;
    float* y    = ws + Y_OFF;
    float* z    = ws + Z_OFF;
    float* yo   = ws + YO_OFF;
    float* st   = ws + STAT_OFF;

    float* csum1 = st + S_CSUM1; float* csq1 = st + S_CSQ1;
    float* sc1   = st + S_SC1;   float* bi1  = st + S_BI1;
    float* bsum  = st + S_BSUM;  float* bsq  = st + S_BSQ;
    float* mu    = st + S_MU;    float* rs   = st + S_RS;
    float* csum2 = st + S_CSUM2; float* csq2 = st + S_CSQ2;
    float* sc2   = st + S_SC2;   float* bi2  = st + S_BI2;

    const float invN = 1.0f / (float)NPIX;

    zero_stats<<<1, 256, 0, stream>>>(st);
    conv_wmma<<<BB * HH * (WW / STRIP), 256, 0, stream>>>(x, conv_w, out);
    chan_stats<<<COUT * 8, 256, 0, stream>>>(out, csum1, csq1);
    bn_finalize<<<1, COUT, 0, stream>>>(csum1, csq1, bn1_g, bn1_b, invN, sc1, bi1);
    xm_kernel<<<BB * COUT * 2, 256, 0, stream>>>(out, sc1, bi1, xm, bsum, bsq);
    ln_finalize<<<1, BB, 0, stream>>>(bsum, bsq, mu, rs);
    sgemm_wmma<<<BB * 128, 256, 0, stream>>>(xm, mu, rs, f);
    softmax_kernel<<<BB * HH, 256, 0, stream>>>(f);
    gemm128_wmma<<<BB * 32, 256, 0, stream>>>(g_w, xm, g_b, mu, rs, 1, y);
    zgemm_wmma<<<BB * 32, 256, 0, stream>>>(y, f, z);
    gemm128_wmma<<<BB * 32, 256, 0, stream>>>(out_w, z, out_b, mu, rs, 0, yo);
    residual_stats<<<BB * COUT * 8, 256, 0, stream>>>(out, sc1, bi1, yo, csum2, csq2);
    bn_finalize<<<1, COUT, 0, stream>>>(csum2, csq2, bn2_g, bn2_b, invN, sc2, bi2);
    silu_apply<<<(BB * COUT * HH * WW) / 256, 256, 0, stream>>>(out, sc2, bi2);
}